// DetectionLoss_31069793419473
// MI455X (gfx1250) — compile-verified
//
#include <hip/hip_runtime.h>
#include <hip/hip_bf16.h>

// ---------------------------------------------------------------------------
// Detection loss for MI455X (gfx1250, wave32).
//   pred   : (32, 24, 160, 160) fp32   d_in[0]
//   targets: (512, 6)          fp32    d_in[1]
//   out    : scalar fp32
//
// Dense pass reads ONLY the 3 conf channels per batch (96 planes, 9.83 MB),
// vectorized float4 (GLOBAL_LOAD_B128). Sparse pass handles the 1536 masked
// cells. Cross-lane reductions use V_WMMA_F32_16X16X4_F32 (ones-B trick).
// No atomics: block partials land in d_ws slots; a combine kernel reduces.
// ---------------------------------------------------------------------------

#define Bsz   32
#define Anch  3
#define Hdim  160
#define Wdim  160
#define HW    (Hdim * Wdim)          // 25600
#define NCLS  3
#define NPLANES (Bsz * Anch)         // 96 conf planes
#define F4_PER_PLANE (HW / 4)        // 6400
#define TOTAL_F4 (NPLANES * F4_PER_PLANE)  // 614400
#define CONF_DENOM 2457600.0f        // B*A*H*W

#define K1_BLOCKS 600
#define K1_THREADS 256
#define K1_ITERS 4                   // 600*256*4 == 614400 exactly

#define NTGT 512
#define K2_BLOCKS 6
#define K2_THREADS 256               // 6*256 == 1536 == NTGT*Anch

// ws layout (floats): [0..599] kernel1 block partials
//                     [600 + blk*4 + {0:box,1:corr,2:cls,3:cnt}] kernel2 (6 blocks)
#define WS_K2_BASE 600

typedef float v2f __attribute__((ext_vector_type(2)));
typedef float v8f __attribute__((ext_vector_type(8)));

__device__ __forceinline__ float fast_sigmoid(float x) {
    return __builtin_amdgcn_rcpf(1.0f + __expf(-x));
}

// Wave32 all-lane sum via V_WMMA_F32_16X16X4_F32 with B == ones:
// A[m][0] = acc(lane m), A[m][2] = acc(lane 16+m)  ->  D[m][n] = acc[m]+acc[m+16]
// sum over 8 D VGPRs gives half-wave total; xor-16 shuffle completes it.
// Requires EXEC == all ones (callers guarantee full waves, no divergence).
__device__ __forceinline__ float wave_sum_wmma(float acc) {
    v2f a; a[0] = acc;  a[1] = 0.0f;
    v2f b; b[0] = 1.0f; b[1] = 1.0f;
    v8f c = {};
    v8f d = __builtin_amdgcn_wmma_f32_16x16x4_f32(
        false, a, false, b, (short)0, c, false, false);
    float p = d[0] + d[1] + d[2] + d[3] + d[4] + d[5] + d[6] + d[7];
    p += __shfl_xor(p, 16, 32);
    return p;  // valid in every lane
}

// -------- Kernel 1: dense sum of sigmoid(conf)^2 over the 96 conf planes ----
__global__ __launch_bounds__(K1_THREADS)
void k1_conf_stream(const float* __restrict__ pred, float* __restrict__ ws) {
    const float4* __restrict__ p4 = reinterpret_cast<const float4*>(pred);
    const int tid    = threadIdx.x;
    const int gid    = blockIdx.x * K1_THREADS + tid;
    const int stride = K1_BLOCKS * K1_THREADS;

    float acc = 0.0f;
#pragma unroll
    for (int k = 0; k < K1_ITERS; ++k) {
        int e     = gid + k * stride;          // < TOTAL_F4
        int plane = e / F4_PER_PLANE;
        int off   = e - plane * F4_PER_PLANE;
        int bb    = plane / Anch;
        int aa    = plane - bb * Anch;
        int ch    = bb * (Anch * 8) + aa * 8 + 4;   // conf channel
        float4 v  = p4[ch * F4_PER_PLANE + off];
        float s0 = fast_sigmoid(v.x);
        float s1 = fast_sigmoid(v.y);
        float s2 = fast_sigmoid(v.z);
        float s3 = fast_sigmoid(v.w);
        acc += s0 * s0 + s1 * s1 + s2 * s2 + s3 * s3;
    }

    // wave -> block reduction
    __shared__ float lds[8];
    float wsum = wave_sum_wmma(acc);
    const int lane = tid & 31, wave = tid >> 5;
    if (lane == 0) lds[wave] = wsum;
    __syncthreads();
    if (tid < 32) {                       // full wave 0, EXEC all ones
        float v = (tid < 8) ? lds[tid] : 0.0f;
        float t = wave_sum_wmma(v);
        if (tid == 0) ws[blockIdx.x] = t;
    }
}

// -------- Kernel 2: sparse target terms (box, conf correction, cls, count) --
__global__ __launch_bounds__(K2_THREADS)
void k2_targets(const float* __restrict__ pred,
                const float* __restrict__ targets,
                float* __restrict__ ws) {
    const int t  = blockIdx.x * K2_THREADS + threadIdx.x;  // 0..1535
    const int i  = t / Anch;          // target index
    const int aa = t - i * Anch;      // anchor

    float box = 0.0f, corr = 0.0f, cls = 0.0f, cnt = 0.0f;

    const float* tg = targets + i * 6;
    const int   bb = (int)tg[0];
    const int   cc = (int)tg[1];
    const float gx = tg[2], gy = tg[3], tw = tg[4], th = tg[5];
    const int gxi = (int)(gx * (float)Wdim);
    const int gyi = (int)(gy * (float)Hdim);
    const bool valid = (gxi >= 0) & (gxi < Wdim) & (gyi >= 0) & (gyi < Hdim) &
                       (bb >= 0) & (bb < Bsz);

    if (valid) {
        const long base = ((long)(bb * (Anch * 8) + aa * 8) * Hdim + gyi) * Wdim + gxi;
        const float p0 = pred[base];
        const float p1 = pred[base + 1L * HW];
        const float p2 = pred[base + 2L * HW];
        const float p3 = pred[base + 3L * HW];
        const float p4 = pred[base + 4L * HW];
        const float p5 = pred[base + 5L * HW];
        const float p6 = pred[base + 6L * HW];
        const float p7 = pred[base + 7L * HW];

        const float dx = fast_sigmoid(p0) - gx;
        const float dy = fast_sigmoid(p1) - gy;
        const float dw = __expf(p2) - tw;
        const float dh = __expf(p3) - th;
        box = dx * dx + dy * dy + dw * dw + dh * dh;

        // masked cell: (s-1)^2 replaces the s^2 already counted in kernel 1
        corr = 1.0f - 2.0f * fast_sigmoid(p4);

        const bool chit = (cc >= 0) & (cc < NCLS);
        const float c0 = fast_sigmoid(p5) - ((chit && cc == 0) ? 1.0f : 0.0f);
        const float c1 = fast_sigmoid(p6) - ((chit && cc == 1) ? 1.0f : 0.0f);
        const float c2 = fast_sigmoid(p7) - ((chit && cc == 2) ? 1.0f : 0.0f);
        cls = c0 * c0 + c1 * c1 + c2 * c2;

        cnt = 1.0f;
    }

    __shared__ float lds[8 * 4];
    const int lane = threadIdx.x & 31, wave = threadIdx.x >> 5;
    const float rb = wave_sum_wmma(box);
    const float rr = wave_sum_wmma(corr);
    const float rc = wave_sum_wmma(cls);
    const float rn = wave_sum_wmma(cnt);
    if (lane == 0) {
        lds[wave * 4 + 0] = rb; lds[wave * 4 + 1] = rr;
        lds[wave * 4 + 2] = rc; lds[wave * 4 + 3] = rn;
    }
    __syncthreads();
    if (threadIdx.x < 4) {
        float s = 0.0f;
        for (int w = 0; w < 8; ++w) s += lds[w * 4 + threadIdx.x];
        ws[WS_K2_BASE + blockIdx.x * 4 + threadIdx.x] = s;
    }
}

// -------- Kernel 3: combine block partials into the final scalar ------------
__global__ __launch_bounds__(32)
void k3_combine(const float* __restrict__ ws, float* __restrict__ out) {
    float s = 0.0f;
    for (int j = threadIdx.x; j < K1_BLOCKS; j += 32) s += ws[j];
    const float S0 = wave_sum_wmma(s);     // dense sigmoid(conf)^2 sum

    if (threadIdx.x == 0) {
        float box = 0.0f, corr = 0.0f, cls = 0.0f, n = 0.0f;
        for (int blk = 0; blk < K2_BLOCKS; ++blk) {
            box  += ws[WS_K2_BASE + blk * 4 + 0];
            corr += ws[WS_K2_BASE + blk * 4 + 1];
            cls  += ws[WS_K2_BASE + blk * 4 + 2];
            n    += ws[WS_K2_BASE + blk * 4 + 3];
        }
        const float loss_conf = (S0 + corr) / CONF_DENOM;
        const float loss_box  = box / (n * 4.0f);
        const float loss_cls  = cls / (n * (float)NCLS);
        out[0] = 5.0f * loss_box + loss_conf + loss_cls;
    }
}

extern "C" void kernel_launch(void* const* d_in, const int* in_sizes, int n_in,
                              void* d_out, int out_size, void* d_ws, size_t ws_size,
                              hipStream_t stream) {
    const float* pred    = (const float*)d_in[0];
    const float* targets = (const float*)d_in[1];
    float* ws  = (float*)d_ws;
    float* out = (float*)d_out;

    k1_conf_stream<<<K1_BLOCKS, K1_THREADS, 0, stream>>>(pred, ws);
    k2_targets<<<K2_BLOCKS, K2_THREADS, 0, stream>>>(pred, targets, ws);
    k3_combine<<<1, 32, 0, stream>>>(ws, out);
}